// LightGCN_88244398064122
// MI455X (gfx1250) — compile-verified
//
#include <hip/hip_runtime.h>

// ---------------------------------------------------------------------------
// LightGCN propagation for MI455X (gfx1250).
// Memory/atomic-bound graph gather-scatter; working set (~135MB) fits in the
// 192MB global L2, so the design goal is coalesced 256B row accesses +
// L2-resident f32 atomics, with the CDNA5 async global->LDS data mover
// (ASYNCcnt / s_wait_asynccnt) keeping 8 row-gathers in flight per wave.
// ---------------------------------------------------------------------------

#ifndef EPW
#define EPW 8              // edges per wave in the scatter kernel (async depth)
#endif

typedef int   v2i __attribute__((vector_size(8)));               // 2 x i32
typedef float v2f __attribute__((vector_size(8)));               // 2 x f32
typedef __attribute__((address_space(1))) v2i* gbl_v2i_ptr;      // global (AS1)
typedef __attribute__((address_space(3))) v2i* lds_v2i_ptr;      // LDS (AS3)
typedef __attribute__((address_space(3))) v2f* lds_v2f_ptr;
typedef __attribute__((address_space(3))) volatile v2f* lds_v2f_vptr;

#if __has_builtin(__builtin_amdgcn_global_load_async_to_lds_b64)
#define HAVE_ASYNC_LDS 1
#else
#define HAVE_ASYNC_LDS 0
#endif

template <int Nw>
__device__ __forceinline__ void async_wait() {
#if HAVE_ASYNC_LDS
#if __has_builtin(__builtin_amdgcn_s_wait_asynccnt)
  __builtin_amdgcn_s_wait_asynccnt(Nw);
#else
  asm volatile("s_wait_asynccnt %0" ::"i"(Nw) : "memory");
#endif
#endif
}

// Compile-time drain loop: step K waits for ASYNCcnt <= N-1-K (an ICE) then
// processes slot K.  Async loads return in order, so this releases exactly
// slot K while K+1..N-1 stay in flight.
template <int K, int N, typename F>
__device__ __forceinline__ void drain(F&& f) {
  if constexpr (K < N) {
    async_wait<N - 1 - K>();
    f(K);
    drain<K + 1, N>(f);
  }
}

// Scatter: xout[row[e]] += xin[col[e]] * rinv[row]*rinv[col]
// One wave per EPW consecutive edges. Lane l owns dims [2l, 2l+2).
// All EPW source rows (256B each, fully coalesced across 32 lanes) are
// launched as async global->LDS loads up front; the drain loop waits with
// descending ASYNCcnt immediates so each edge's scatter overlaps the
// remaining in-flight gathers.  Scatter = global_atomic_add_f32 (L2-resident).
__global__ __launch_bounds__(256) void k_scatter(
    const int* __restrict__ row, const int* __restrict__ col,
    const float* __restrict__ rinv,
    const float* __restrict__ xin, float* __restrict__ xout, int E) {
  __shared__ float lds[8][EPW][64];        // 8 waves/block * EPW slots * 256B = 16KB
  const int wave = threadIdx.x >> 5;
  const int lane = threadIdx.x & 31;
  const long long ebase = ((long long)blockIdx.x * 8 + wave) * (long long)EPW;
  if (ebase >= (long long)E) return;
  const int nE = (int)(((long long)E - ebase) < (long long)EPW ? ((long long)E - ebase)
                                                               : (long long)EPW);

  auto stage = [&](int k) {
    const int c = col[ebase + k];          // wave-uniform
    const float* src = xin + ((size_t)c << 6) + (size_t)(lane << 1);
#if HAVE_ASYNC_LDS
    __builtin_amdgcn_global_load_async_to_lds_b64(
        (gbl_v2i_ptr)src, (lds_v2i_ptr)&lds[wave][k][lane << 1],
        /*offset=*/0, /*cpol=*/0);
#else
    *(lds_v2f_ptr)&lds[wave][k][lane << 1] = *(const v2f*)src;
#endif
  };

  auto process = [&](int k) {
    const int r = row[ebase + k];
    const int c = col[ebase + k];
    const float w = rinv[r] * rinv[c];
    const v2f ab = *(lds_v2f_vptr)&lds[wave][k][lane << 1];  // ds_load_b64
    float* dst = xout + ((size_t)r << 6) + (size_t)(lane << 1);
    atomicAdd(dst + 0, ab[0] * w);
    atomicAdd(dst + 1, ab[1] * w);
  };

  if (nE == EPW) {
    // Full batch: issue all EPW async gathers, drain with descending waits.
#pragma unroll
    for (int k = 0; k < EPW; ++k) stage(k);
    drain<0, EPW>(process);
  } else {
    for (int k = 0; k < nE; ++k) stage(k);
    async_wait<0>();
    for (int k = 0; k < nE; ++k) process(k);
  }
}

// x0 = acc = concat(user_emb, item_emb), vectorized float4
__global__ void k_init(const float4* __restrict__ ue, const float4* __restrict__ ie,
                       float4* __restrict__ x0, float4* __restrict__ acc,
                       int u4, int n4) {
  int i = blockIdx.x * blockDim.x + threadIdx.x;
  if (i >= n4) return;
  float4 v = (i < u4) ? ue[i] : ie[i - u4];
  x0[i] = v;
  acc[i] = v;
}

// deg[row[e]] += 1  (u32 atomics; deg buffer pre-zeroed via memsetAsync)
__global__ void k_deg(const int* __restrict__ row, unsigned* __restrict__ deg, int E) {
  int e = blockIdx.x * blockDim.x + threadIdx.x;
  if (e < E) atomicAdd(&deg[row[e]], 1u);
}

// rinv[i] = rsqrt(max(deg[i], 1))
__global__ void k_rinv(const unsigned* __restrict__ deg, float* __restrict__ rinv, int N) {
  int i = blockIdx.x * blockDim.x + threadIdx.x;
  if (i >= N) return;
  unsigned d = deg[i];
  rinv[i] = rsqrtf((float)(d < 1u ? 1u : d));
}

// acc = (acc + x) * scale   (scale = 1 for inner layers, 0.25 on the last)
__global__ void k_axpy(float4* __restrict__ acc, const float4* __restrict__ x,
                       float scale, int n4) {
  int i = blockIdx.x * blockDim.x + threadIdx.x;
  if (i >= n4) return;
  float4 a = acc[i];
  const float4 b = x[i];
  a.x = (a.x + b.x) * scale;
  a.y = (a.y + b.y) * scale;
  a.z = (a.z + b.z) * scale;
  a.w = (a.w + b.w) * scale;
  acc[i] = a;
}

extern "C" void kernel_launch(void* const* d_in, const int* in_sizes, int n_in,
                              void* d_out, int out_size, void* d_ws, size_t ws_size,
                              hipStream_t stream) {
  const float* ue = (const float*)d_in[0];        // [100000, 64] f32
  const float* ie = (const float*)d_in[1];        // [50000, 64]  f32
  const int*   ei = (const int*)d_in[2];          // [2, E]       i32

  const int uElems = in_sizes[0];
  const int iElems = in_sizes[1];
  const int E = in_sizes[2] / 2;
  const int N = (uElems + iElems) / 64;
  const int nElem = N * 64;

  const int* row = ei;
  const int* col = ei + E;
  float* out = (float*)d_out;                     // [N, 64] f32 (users then items)

  // Workspace layout (256B aligned): deg | rinv | x0 | x1   (~78 MB total)
  char* ws = (char*)d_ws;
  size_t off = 0;
  auto walloc = [&](size_t bytes) -> void* {
    void* p = ws + off;
    off = (off + bytes + 255) & ~(size_t)255;
    return p;
  };
  unsigned* deg  = (unsigned*)walloc((size_t)N * 4);
  float*    rinv = (float*)   walloc((size_t)N * 4);
  float*    x0   = (float*)   walloc((size_t)nElem * 4);
  float*    x1   = (float*)   walloc((size_t)nElem * 4);

  const int n4 = nElem / 4;
  (void)hipMemsetAsync(deg, 0, (size_t)N * 4, stream);
  k_init<<<(n4 + 255) / 256, 256, 0, stream>>>((const float4*)ue, (const float4*)ie,
                                               (float4*)x0, (float4*)out,
                                               uElems / 4, n4);
  k_deg<<<(E + 255) / 256, 256, 0, stream>>>(row, deg, E);
  k_rinv<<<(N + 255) / 256, 256, 0, stream>>>(deg, rinv, N);

  const int waves = (E + EPW - 1) / EPW;
  const int sblocks = (waves + 7) / 8;            // 8 waves per 256-thread block

  float* xc = x0;
  float* xn = x1;
  for (int layer = 0; layer < 3; ++layer) {
    (void)hipMemsetAsync(xn, 0, (size_t)nElem * 4, stream);
    k_scatter<<<sblocks, 256, 0, stream>>>(row, col, rinv, xc, xn, E);
    const float sc = (layer == 2) ? 0.25f : 1.0f;
    k_axpy<<<(n4 + 255) / 256, 256, 0, stream>>>((float4*)out, (const float4*)xn, sc, n4);
    float* t = xc; xc = xn; xn = t;
  }
}